// MyGraphConv_69045894250720
// MI455X (gfx1250) — compile-verified
//
#include <hip/hip_runtime.h>
#include <hip/hip_bf16.h>

typedef __attribute__((ext_vector_type(2))) float v2f;
typedef __attribute__((ext_vector_type(8))) float v8f;

#define FEATS 128
#define LDS_STRIDE 132   // 128 + 4 pad: column reads hit distinct banks (64 banks x 4B)

// Hardware fp32 global atomic add (global_atomic_add_f32, no CAS loop).
__device__ __forceinline__ void hw_atomic_add(float* p, float v) {
#if defined(__HIP_DEVICE_COMPILE__)
    unsafeAtomicAdd(p, v);
#else
    atomicAdd(p, v);
#endif
}

// ---------------------------------------------------------------------------
// Kernel 1: zero the workspace accumulators (summed[N*128] ++ deg[N])
// ---------------------------------------------------------------------------
__global__ void gc_zero_kernel(float* __restrict__ p, int n) {
    int i = blockIdx.x * blockDim.x + threadIdx.x;
    if (i < n) p[i] = 0.0f;
}

// ---------------------------------------------------------------------------
// Kernel 2: edge scatter. One wave32 per edge: lane l handles feats 4l..4l+3.
// Gather x[src] as float4 (coalesced 512B row), scatter-add into summed[dst]
// with hardware f32 far atomics. Accumulator (25.8MB) is L2-resident (192MB
// L2), so the RMW traffic never touches HBM.
// ---------------------------------------------------------------------------
__global__ void __launch_bounds__(256)
gc_scatter_kernel(const float* __restrict__ x,
                  const int*   __restrict__ src,
                  const int*   __restrict__ dst,
                  float*       __restrict__ summed,
                  float*       __restrict__ deg,
                  int E) {
    int gid  = blockIdx.x * blockDim.x + threadIdx.x;
    int e    = gid >> 5;
    int lane = gid & 31;
    if (e >= E) return;

    int s = src[e];
    int d = dst[e];

    const float4 v = ((const float4*)(x + (size_t)s * FEATS))[lane];
    float* o = summed + (size_t)d * FEATS + lane * 4;
    hw_atomic_add(o + 0, v.x);
    hw_atomic_add(o + 1, v.y);
    hw_atomic_add(o + 2, v.z);
    hw_atomic_add(o + 3, v.w);
    if (lane == 0) hw_atomic_add(deg + d, 1.0f);
}

// ---------------------------------------------------------------------------
// Kernel 3: mean-normalize + GEMM via V_WMMA_F32_16X16X4_F32.
// Block = 256 threads = 8 waves, one 16-row M-tile per block.
// Stage h = summed/max(deg,1) tile (16x128) in LDS; wave w computes the
// 16x16 output tile at N-tile w, accumulating K=128 in 32 wmma steps.
// ---------------------------------------------------------------------------
__global__ void __launch_bounds__(256, 2)
gc_gemm_kernel(const float* __restrict__ summed,
               const float* __restrict__ deg,
               const float* __restrict__ Wm,
               const float* __restrict__ bias,
               float*       __restrict__ out) {
    __shared__ float hs[16 * LDS_STRIDE];

    const int m0  = blockIdx.x * 16;
    const int tid = threadIdx.x;

    // Stage the normalized 16x128 h tile: 2048 elems / 256 threads = 8 each.
    #pragma unroll
    for (int i = 0; i < 8; ++i) {
        int idx = tid + i * 256;
        int row = idx >> 7;      // /128
        int col = idx & 127;
        float dg    = deg[m0 + row];
        float scale = 1.0f / fmaxf(dg, 1.0f);
        hs[row * LDS_STRIDE + col] =
            summed[(size_t)(m0 + row) * FEATS + col] * scale;
    }
    __syncthreads();

    const int wave = tid >> 5;
    const int lane = tid & 31;
    const int n0   = wave * 16;       // this wave's N-tile
    const int half = lane >> 4;       // 0: K-pair {0,1}, 1: K-pair {2,3}
    const int l    = lane & 15;       // M index for A, N index for B/C

    // A fragment source: row M=l, K offset (half*2)
    const float* hrow = hs + l * LDS_STRIDE + half * 2;

    v8f c = {};
    #pragma unroll 4
    for (int k0 = 0; k0 < FEATS; k0 += 4) {
        v2f a;
        a.x = hrow[k0 + 0];
        a.y = hrow[k0 + 1];
        v2f bfr;
        bfr.x = Wm[(size_t)(k0 + half * 2 + 0) * FEATS + n0 + l];
        bfr.y = Wm[(size_t)(k0 + half * 2 + 1) * FEATS + n0 + l];
        // D = A(16x4) x B(4x16) + C   (full fp32 WMMA)
        c = __builtin_amdgcn_wmma_f32_16x16x4_f32(
                /*neg_a=*/false, a, /*neg_b=*/false, bfr,
                /*c_mod=*/(short)0, c, /*reuse_a=*/false, /*reuse_b=*/false);
    }

    // C/D layout: VGPR r -> M = r + 8*half (lanes 0-15: M=r, 16-31: M=r+8), N = l
    const float bv = bias[n0 + l];
    #pragma unroll
    for (int r = 0; r < 8; ++r) {
        int m = m0 + r + half * 8;
        out[(size_t)m * FEATS + n0 + l] = c[r] + bv;
    }
}

// ---------------------------------------------------------------------------
// Launch
// ---------------------------------------------------------------------------
extern "C" void kernel_launch(void* const* d_in, const int* in_sizes, int n_in,
                              void* d_out, int out_size, void* d_ws, size_t ws_size,
                              hipStream_t stream) {
    const float* x   = (const float*)d_in[0];
    const int*   src = (const int*)  d_in[1];
    const int*   dst = (const int*)  d_in[2];
    const float* Wm  = (const float*)d_in[3];
    const float* bv  = (const float*)d_in[4];
    float*       out = (float*)d_out;

    const int N = in_sizes[0] / FEATS;   // 50000
    const int E = in_sizes[1];           // 800000

    float* summed = (float*)d_ws;                 // N * 128 floats
    float* deg    = summed + (size_t)N * FEATS;   // N floats

    // 1) zero accumulators
    int zn = N * FEATS + N;
    gc_zero_kernel<<<(zn + 255) / 256, 256, 0, stream>>>(summed, zn);

    // 2) edge scatter: one wave32 per edge
    long threads = (long)E * 32;
    gc_scatter_kernel<<<(unsigned)((threads + 255) / 256), 256, 0, stream>>>(
        x, src, dst, summed, deg, E);

    // 3) normalize + GEMM (N divisible by 16: 50000 = 3125 * 16)
    gc_gemm_kernel<<<N / 16, 256, 0, stream>>>(summed, deg, Wm, bv, out);
}